// DeLanJacobianNet_inverse_66477503808141
// MI455X (gfx1250) — compile-verified
//
#include <hip/hip_runtime.h>

typedef __attribute__((ext_vector_type(2))) float v2f;
typedef __attribute__((ext_vector_type(8))) float v8f;

namespace {
constexpr int S   = 44;        // LDS column stride (floats) - bank-conflict padding
constexpr int NW  = 2;         // waves per block
constexpr int BLK = NW * 32;   // 64 threads

// ---- weight LDS offsets (floats) ----
constexpr int O_JPW1 = 0;
constexpr int O_JPB1 = O_JPW1 + 7 * 30;
constexpr int O_JPW2 = O_JPB1 + 30;
constexpr int O_JPB2 = O_JPW2 + 30 * 21;
constexpr int O_JOW1 = O_JPB2 + 21;
constexpr int O_JOB1 = O_JOW1 + 7 * 7 * 30;
constexpr int O_JOW2 = O_JOB1 + 7 * 30;
constexpr int O_JOB2 = O_JOW2 + 7 * 30 * 21;
constexpr int O_GW1  = O_JOB2 + 7 * 21;
constexpr int O_GB1  = O_GW1 + 7 * 20;
constexpr int O_GW2  = O_GB1 + 20;
constexpr int O_GB2  = O_GW2 + 20 * 7;
constexpr int O_WP   = O_GB2 + 7;       // 49 floats (Wp)
constexpr int O_IM   = O_WP + 49;       // 63 floats (I_mat 7x3x3)
constexpr int WGT    = O_IM + 63;

// ---- per-wave activation scratch offsets (floats) ----
constexpr int A_Q  = 0;
constexpr int A_U  = A_Q  + 7  * S;
constexpr int A_H  = A_U  + 7  * S;   // sin(Z), also reused for g-net hidden
constexpr int A_CZ = A_H  + 30 * S;   // cos(Z)
constexpr int A_DH = A_CZ + 30 * S;   // dh = cos(Z)*dZ
constexpr int A_E  = A_DH + 30 * S;   // e = gs * cos(Z)
constexpr int A_JO = A_E  + 30 * S;   // Jp/Jo output (21 rows)
constexpr int A_DJ = A_JO + 21 * S;
constexpr int A_G  = A_DJ + 21 * S;   // cotangent wrt Jp/Jo
constexpr int A_C1 = A_G  + 21 * S;   // c1 accumulator (+ g net), 7 rows
constexpr int WAVE = A_C1 + 7  * S;

constexpr int SH_FLOATS = WGT + NW * WAVE;
} // namespace

// upper-triangle read for symmetric 7x7 stored as 28-entry triangle
__device__ __forceinline__ float symread(const float* T, int i, int j) {
  return (i <= j) ? T[i * 7 - (i * (i - 1)) / 2 + (j - i)]
                  : T[j * 7 - (j * (j - 1)) / 2 + (i - j)];
}

__device__ __forceinline__ void cpy(float* dst, const float* src, int n, int tid) {
  for (int i = tid; i < n; i += BLK) dst[i] = src[i];
}

// init C tiles with broadcast bias (or zero); bias==nullptr folds away after inlining
template <int M>
__device__ __forceinline__ void cinit(const float* bias, int lane, v8f* C) {
  constexpr int MT = (M + 15) / 16;
  const int rr = (lane >> 4) * 8;
#pragma unroll
  for (int mt = 0; mt < MT; ++mt) {
#pragma unroll
    for (int r = 0; r < 8; ++r) {
      const int m = mt * 16 + rr + r;
      if (bias != nullptr) {
        const float t = bias[m < M ? m : 0];      // unconditional clamped load
        C[mt][r] = (m < M) ? t : 0.0f;            // value select, no exec branch
      } else {
        C[mt][r] = 0.0f;
      }
    }
  }
}

// D = A(MxK) * B(Kx16) + C using V_WMMA_F32_16X16X4_F32.
// A from LDS weights: AT=false -> W stored [k*M+m]; AT=true -> W stored [m*K+k].
// B from LDS activations stored feature-major: Bl[k*S + col], cols 16*half..16*half+15.
// All padded fetches are clamped unconditional LDS loads + value selects.
template <int K, int M, bool AT>
__device__ __forceinline__ void gacc(const float* Wl, const float* Bl,
                                     int half, int lane, v8f* C) {
  constexpr int MT = (M + 15) / 16;
  constexpr int KT = (K + 3) / 4;
  const int mlo = lane & 15;
  const int kh  = (lane >> 4) * 2;       // lanes 16-31 carry K+2, K+3
  const int n   = (lane & 15) + half * 16;
#pragma unroll
  for (int kt = 0; kt < KT; ++kt) {
    const int k0  = kt * 4 + kh;
    const int k1  = k0 + 1;
    const int kc0 = (k0 < K) ? k0 : 0;
    const int kc1 = (k1 < K) ? k1 : 0;
    const float b0 = Bl[kc0 * S + n];
    const float b1 = Bl[kc1 * S + n];
    v2f bv;
    bv.x = (k0 < K) ? b0 : 0.0f;
    bv.y = (k1 < K) ? b1 : 0.0f;
#pragma unroll
    for (int mt = 0; mt < MT; ++mt) {
      const int m  = mt * 16 + mlo;
      const int mc = (m < M) ? m : 0;
      const float a0 = AT ? Wl[mc * K + kc0] : Wl[kc0 * M + mc];
      const float a1 = AT ? Wl[mc * K + kc1] : Wl[kc1 * M + mc];
      v2f av;
      av.x = (m < M && k0 < K) ? a0 : 0.0f;
      av.y = (m < M && k1 < K) ? a1 : 0.0f;
      C[mt] = __builtin_amdgcn_wmma_f32_16x16x4_f32(false, av, false, bv, (short)0,
                                                    C[mt], false, false);
    }
  }
}

// store D tiles with elementwise post-op f(row, col, value)
template <int M, typename F>
__device__ __forceinline__ void dstore(v8f* C, int half, int lane, F f) {
  constexpr int MT = (M + 15) / 16;
  const int col = (lane & 15) + half * 16;
  const int rr  = (lane >> 4) * 8;
#pragma unroll
  for (int mt = 0; mt < MT; ++mt) {
#pragma unroll
    for (int r = 0; r < 8; ++r) {
      const int m = mt * 16 + rr + r;
      if (m < M) f(m, col, C[mt][r]);
    }
  }
}

// forward + JVP pass of one 7->30->21 net (Jp or Jo_n).
// __forceinline__ is essential: keeps the __shared__ provenance visible so all
// accesses lower to ds_load/ds_store (addrspace 3), not flat.
__device__ __forceinline__ void net_fwd(
    const float* W1, const float* B1, const float* W2, const float* B2,
    const float* Q, const float* U,
    float* h, float* cz, float* dh, float* Jo, float* dJo, int lane) {
  for (int half = 0; half < 2; ++half) {             // Z = W1^T q + b1 ; h=sin, cz=cos
    v8f C[2]; cinit<30>(B1, lane, C);
    gacc<7, 30, false>(W1, Q, half, lane, C);
    dstore<30>(C, half, lane, [&](int m, int col, float v) {
      // |z| is small here; hardware sin/cos path (TRANS ops, co-execute with WMMA)
      float sv, cv; __sincosf(v, &sv, &cv);
      h[m * S + col] = sv; cz[m * S + col] = cv;
    });
  }
  for (int half = 0; half < 2; ++half) {             // dZ = W1^T u ; dh = cz*dZ
    v8f C[2]; cinit<30>(nullptr, lane, C);
    gacc<7, 30, false>(W1, U, half, lane, C);
    dstore<30>(C, half, lane, [&](int m, int col, float v) {
      dh[m * S + col] = cz[m * S + col] * v;
    });
  }
  for (int half = 0; half < 2; ++half) {             // Jo = W2^T h + b2
    v8f C[2]; cinit<21>(B2, lane, C);
    gacc<30, 21, false>(W2, h, half, lane, C);
    dstore<21>(C, half, lane, [&](int m, int col, float v) { Jo[m * S + col] = v; });
  }
  for (int half = 0; half < 2; ++half) {             // dJo = W2^T dh
    v8f C[2]; cinit<21>(nullptr, lane, C);
    gacc<30, 21, false>(W2, dh, half, lane, C);
    dstore<21>(C, half, lane, [&](int m, int col, float v) { dJo[m * S + col] = v; });
  }
}

// reverse pass: gs = W2 @ G ; e = gs*cos(Z) ; c1 += W1 @ e
__device__ __forceinline__ void net_bwd(
    const float* W1, const float* W2,
    const float* G, const float* cz, float* e, float* c1, int lane) {
  for (int half = 0; half < 2; ++half) {
    v8f C[2]; cinit<30>(nullptr, lane, C);
    gacc<21, 30, true>(W2, G, half, lane, C);
    dstore<30>(C, half, lane, [&](int m, int col, float v) {
      e[m * S + col] = cz[m * S + col] * v;
    });
  }
  for (int half = 0; half < 2; ++half) {
    v8f C[1]; cinit<7>(nullptr, lane, C);
    gacc<30, 7, true>(W1, e, half, lane, C);
    dstore<7>(C, half, lane, [&](int m, int col, float v) { c1[m * S + col] += v; });
  }
}

// precompute Wp (7x7) and I_mat (7x3x3) into d_ws
__global__ void delan_prep(const float* __restrict__ m_param,
                           const float* __restrict__ inertia,
                           float* __restrict__ ws) {
  if (threadIdx.x != 0 || blockIdx.x != 0) return;
  float mc[7], rev[7];
  for (int i = 0; i < 7; ++i) mc[i] = fmaxf(m_param[i], 1e-6f);
  rev[6] = mc[6];
  for (int i = 5; i >= 0; --i) rev[i] = rev[i + 1] + mc[i];
  for (int i = 0; i < 7; ++i)
    for (int j = 0; j < 7; ++j) ws[i * 7 + j] = rev[i > j ? i : j];
  for (int n = 0; n < 7; ++n) {
    const float* t = inertia + n * 6;   // tril order (0,0)(1,0)(1,1)(2,0)(2,1)(2,2)
    float L[3][3] = {{t[0], 0.f, 0.f}, {t[1], t[2], 0.f}, {t[3], t[4], t[5]}};
    for (int a = 0; a < 3; ++a)
      for (int c = 0; c < 3; ++c) {
        float s = 0.f;
        for (int bb = 0; bb < 3; ++bb) s += L[a][bb] * L[c][bb];
        ws[49 + n * 9 + a * 3 + c] = s;
      }
  }
}

__global__ __launch_bounds__(BLK)
void delan_main(const float* __restrict__ x,
                const float* JpW1, const float* JpB1, const float* JpW2, const float* JpB2,
                const float* JoW1, const float* JoB1, const float* JoW2, const float* JoB2,
                const float* gW1,  const float* gB1,  const float* gW2,  const float* gB2,
                const float* __restrict__ ws, float* __restrict__ out) {
  extern __shared__ float sm[];
  const int tid  = threadIdx.x;
  const int lane = tid & 31;
  const int wv   = tid >> 5;
  float* W = sm;

  cpy(W + O_JPW1, JpW1, 7 * 30, tid);
  cpy(W + O_JPB1, JpB1, 30, tid);
  cpy(W + O_JPW2, JpW2, 30 * 21, tid);
  cpy(W + O_JPB2, JpB2, 21, tid);
  cpy(W + O_JOW1, JoW1, 7 * 7 * 30, tid);
  cpy(W + O_JOB1, JoB1, 7 * 30, tid);
  cpy(W + O_JOW2, JoW2, 7 * 30 * 21, tid);
  cpy(W + O_JOB2, JoB2, 7 * 21, tid);
  cpy(W + O_GW1,  gW1, 7 * 20, tid);
  cpy(W + O_GB1,  gB1, 20, tid);
  cpy(W + O_GW2,  gW2, 20 * 7, tid);
  cpy(W + O_GB2,  gB2, 7, tid);
  cpy(W + O_WP,   ws, 112, tid);    // Wp(49) + I_mat(63), contiguous in ws
  __syncthreads();

  float* R  = sm + WGT + wv * WAVE;
  float* Q  = R + A_Q;  float* U  = R + A_U;  float* Hb = R + A_H;
  float* CZ = R + A_CZ; float* Dh = R + A_DH; float* E  = R + A_E;
  float* JO = R + A_JO; float* DJ = R + A_DJ; float* G  = R + A_G;
  float* C1 = R + A_C1;

  const int b = blockIdx.x * BLK + tid;     // one batch element per lane
  const float* xb = x + b * 21;
  float q[7], u[7], qd[7];
#pragma unroll
  for (int i = 0; i < 7; ++i) { q[i] = xb[i]; u[i] = xb[7 + i]; qd[i] = xb[14 + i]; }
#pragma unroll
  for (int i = 0; i < 7; ++i) { Q[i * S + lane] = q[i]; U[i * S + lane] = u[i]; C1[i * S + lane] = 0.0f; }

  float H[28], dH[28];
#pragma unroll
  for (int t = 0; t < 28; ++t) { H[t] = 0.0f; dH[t] = 0.0f; }

  const float* Wp = W + O_WP;
  const float* IM = W + O_IM;

  // ================= Jp net (Hp, weighted by Wp) =================
  net_fwd(W + O_JPW1, W + O_JPB1, W + O_JPW2, W + O_JPB2, Q, U, Hb, CZ, Dh, JO, DJ, lane);
  {
    float Jp[21], dJp[21];
#pragma unroll
    for (int t = 0; t < 21; ++t) { Jp[t] = JO[t * S + lane]; dJp[t] = DJ[t * S + lane]; }
    int t = 0;
#pragma unroll
    for (int i = 0; i < 7; ++i)
#pragma unroll
      for (int j = i; j < 7; ++j) {
        const float wp = Wp[i * 7 + j];
        float s = 0.f, ds = 0.f;
#pragma unroll
        for (int k = 0; k < 3; ++k) {
          const float a = Jp[k * 7 + i], bb = Jp[k * 7 + j];
          s  += a * bb;
          ds += dJp[k * 7 + i] * bb + a * dJp[k * 7 + j];
        }
        H[t] += wp * s; dH[t] += wp * ds; ++t;
      }
    // G[k,i] = 2*u_i * sum_j Wp[i,j]*Jp[k,j]*u_j
#pragma unroll
    for (int k = 0; k < 3; ++k)
#pragma unroll
      for (int i = 0; i < 7; ++i) {
        float acc = 0.f;
#pragma unroll
        for (int j = 0; j < 7; ++j) acc += Wp[i * 7 + j] * Jp[k * 7 + j] * u[j];
        G[(k * 7 + i) * S + lane] = 2.0f * u[i] * acc;
      }
  }
  net_bwd(W + O_JPW1, W + O_JPW2, G, CZ, E, C1, lane);

  // ================= Jo nets (Ho, weighted by I_mat[n]) =================
  for (int n = 0; n < 7; ++n) {
    const float* W1 = W + O_JOW1 + n * 210;
    const float* B1 = W + O_JOB1 + n * 30;
    const float* W2 = W + O_JOW2 + n * 630;
    const float* B2 = W + O_JOB2 + n * 21;
    net_fwd(W1, B1, W2, B2, Q, U, Hb, CZ, Dh, JO, DJ, lane);
    {
      float Jo[21], dJo[21];
#pragma unroll
      for (int t = 0; t < 21; ++t) { Jo[t] = JO[t * S + lane]; dJo[t] = DJ[t * S + lane]; }
      float I9[9];
#pragma unroll
      for (int a = 0; a < 9; ++a) I9[a] = IM[n * 9 + a];
      float A[21], Ad[21];
#pragma unroll
      for (int k = 0; k < 3; ++k)
#pragma unroll
        for (int j = 0; j < 7; ++j) {
          A[k * 7 + j]  = I9[k*3+0]*Jo[j]  + I9[k*3+1]*Jo[7+j]  + I9[k*3+2]*Jo[14+j];
          Ad[k * 7 + j] = I9[k*3+0]*dJo[j] + I9[k*3+1]*dJo[7+j] + I9[k*3+2]*dJo[14+j];
        }
      int t = 0;
#pragma unroll
      for (int i = 0; i < 7; ++i)
#pragma unroll
        for (int j = i; j < 7; ++j) {
          float s = 0.f, ds = 0.f;
#pragma unroll
          for (int k = 0; k < 3; ++k) {
            s  += Jo[k * 7 + i] * A[k * 7 + j];
            ds += dJo[k * 7 + i] * A[k * 7 + j] + Jo[k * 7 + i] * Ad[k * 7 + j];
          }
          H[t] += s; dH[t] += ds; ++t;
        }
      // G[k,i] = 2*u_i*(I w)_k with w_k = sum_i Jo[k,i] u_i
      float w3[3], Iw[3];
#pragma unroll
      for (int k = 0; k < 3; ++k) {
        float s = 0.f;
#pragma unroll
        for (int i = 0; i < 7; ++i) s += Jo[k * 7 + i] * u[i];
        w3[k] = s;
      }
#pragma unroll
      for (int k = 0; k < 3; ++k) Iw[k] = I9[k*3+0]*w3[0] + I9[k*3+1]*w3[1] + I9[k*3+2]*w3[2];
#pragma unroll
      for (int k = 0; k < 3; ++k)
#pragma unroll
        for (int i = 0; i < 7; ++i) G[(k * 7 + i) * S + lane] = 2.0f * u[i] * Iw[k];
    }
    net_bwd(W1, W2, G, CZ, E, C1, lane);
  }

  // ================= gravity net (adds into c1 accumulator) =================
  for (int half = 0; half < 2; ++half) {
    v8f C[2]; cinit<20>(W + O_GB1, lane, C);
    gacc<7, 20, false>(W + O_GW1, Q, half, lane, C);
    dstore<20>(C, half, lane, [&](int m, int col, float v) { Hb[m * S + col] = __sinf(v); });
  }
  for (int half = 0; half < 2; ++half) {
    v8f C[1]; cinit<7>(W + O_GB2, lane, C);
    gacc<20, 7, false>(W + O_GW2, Hb, half, lane, C);
    dstore<7>(C, half, lane, [&](int m, int col, float v) { C1[m * S + col] += v; });
  }

  // ================= tau = qdd@H + c1 + u@dH + g =================
  float* ob = out + b * 7;
#pragma unroll
  for (int j = 0; j < 7; ++j) {
    float s = C1[j * S + lane];
#pragma unroll
    for (int i = 0; i < 7; ++i)
      s += qd[i] * symread(H, i, j) + u[i] * symread(dH, i, j);
    ob[j] = s;
  }
}

extern "C" void kernel_launch(void* const* d_in, const int* in_sizes, int n_in,
                              void* d_out, int out_size, void* d_ws, size_t ws_size,
                              hipStream_t stream) {
  const float* x     = (const float*)d_in[0];
  const float* m_par = (const float*)d_in[1];
  const float* inert = (const float*)d_in[2];
  const float* JpW1  = (const float*)d_in[3];
  const float* JpB1  = (const float*)d_in[4];
  const float* JpW2  = (const float*)d_in[5];
  const float* JpB2  = (const float*)d_in[6];
  const float* JoW1  = (const float*)d_in[7];
  const float* JoB1  = (const float*)d_in[8];
  const float* JoW2  = (const float*)d_in[9];
  const float* JoB2  = (const float*)d_in[10];
  const float* gW1   = (const float*)d_in[11];
  const float* gB1   = (const float*)d_in[12];
  const float* gW2   = (const float*)d_in[13];
  const float* gB2   = (const float*)d_in[14];
  float* ws  = (float*)d_ws;
  float* out = (float*)d_out;

  const int B = in_sizes[0] / 21;   // 32768

  delan_prep<<<1, 32, 0, stream>>>(m_par, inert, ws);
  delan_main<<<B / BLK, BLK, SH_FLOATS * (int)sizeof(float), stream>>>(
      x, JpW1, JpB1, JpW2, JpB2, JoW1, JoB1, JoW2, JoB2,
      gW1, gB1, gW2, gB2, ws, out);
}